// Attention_12747462935000
// MI455X (gfx1250) — compile-verified
//
#include <hip/hip_runtime.h>
#include <hip/hip_bf16.h>

// Problem constants (match reference)
#define BATCH 2
#define LSEQ  1024
#define DDIM  256
#define UNITS 32
#define JT    144          // shifted band window per 16-row block: j in [i0-64, i0+79]
#define EPSV  1e-7f

typedef __attribute__((ext_vector_type(2))) float v2f;
typedef __attribute__((ext_vector_type(8))) float v8f;

__device__ __forceinline__ int clampi(int v, int lo, int hi) {
    return v < lo ? lo : (v > hi ? hi : v);
}

// -----------------------------------------------------------------------------
// Kernel 1: q = x@Wt, k = x@Wx using V_WMMA_F32_16X16X4_F32.
// One wave (32 threads) per 16 rows of x. 4 accumulators: q cols [0,16), [16,32)
// and k cols [0,16), [16,32). K-loop over D=256 in steps of 4.
// -----------------------------------------------------------------------------
__global__ void __launch_bounds__(32)
qk_proj_kernel(const float* __restrict__ x,
               const float* __restrict__ Wt,
               const float* __restrict__ Wx,
               float* __restrict__ qbuf,
               float* __restrict__ kbuf) {
    const int blk  = blockIdx.x;
    const int b    = blk / (LSEQ / 16);
    const int i0   = (blk % (LSEQ / 16)) * 16;
    const int lane = threadIdx.x;
    const int half = lane >> 4;       // 0: K=0..1,  1: K=2..3  (A);  rows of B likewise
    const int lr   = lane & 15;       // row (A) / column (B,C,D) within tile

    v8f accq0 = {}, accq1 = {}, acck0 = {}, acck1 = {};

    const float* xrow = x + ((size_t)b * LSEQ + i0) * DDIM;

    for (int kk = 0; kk < DDIM; kk += 4) {
        // A fragment: 16x4 f32. lanes 0-15 -> K kk+0/kk+1; lanes 16-31 -> kk+2/kk+3
        v2f a;
        a.x = xrow[(size_t)lr * DDIM + kk + half * 2];
        a.y = xrow[(size_t)lr * DDIM + kk + half * 2 + 1];

        const int brow = kk + half * 2;   // B fragment rows for this half-wave
        v2f bt0, bt1, bx0, bx1;
        bt0.x = Wt[(size_t)brow * UNITS + lr];
        bt0.y = Wt[(size_t)(brow + 1) * UNITS + lr];
        bt1.x = Wt[(size_t)brow * UNITS + 16 + lr];
        bt1.y = Wt[(size_t)(brow + 1) * UNITS + 16 + lr];
        bx0.x = Wx[(size_t)brow * UNITS + lr];
        bx0.y = Wx[(size_t)(brow + 1) * UNITS + lr];
        bx1.x = Wx[(size_t)brow * UNITS + 16 + lr];
        bx1.y = Wx[(size_t)(brow + 1) * UNITS + 16 + lr];

        accq0 = __builtin_amdgcn_wmma_f32_16x16x4_f32(false, a, false, bt0, (short)0, accq0, false, false);
        accq1 = __builtin_amdgcn_wmma_f32_16x16x4_f32(false, a, false, bt1, (short)0, accq1, false, false);
        acck0 = __builtin_amdgcn_wmma_f32_16x16x4_f32(false, a, false, bx0, (short)0, acck0, false, false);
        acck1 = __builtin_amdgcn_wmma_f32_16x16x4_f32(false, a, false, bx1, (short)0, acck1, false, false);
    }

    // C/D layout: VGPR v holds row v (lanes 0-15) / row v+8 (lanes 16-31), col = lr
    for (int v = 0; v < 8; ++v) {
        const int row = v + half * 8;
        const size_t base = ((size_t)b * LSEQ + i0 + row) * UNITS;
        qbuf[base + lr]      = accq0[v];
        qbuf[base + 16 + lr] = accq1[v];
        kbuf[base + lr]      = acck0[v];
        kbuf[base + 16 + lr] = acck1[v];
    }
}

// -----------------------------------------------------------------------------
// Kernel 2: banded emission + softmax + v = a @ x.
// One 128-thread block (4 waves) per (batch, 16-row block).
//   - Stage q rows, k window, bh, Wa in LDS.
//   - Emission: e = Wa . tanh(q_i + k_j + bh) + ba; a = band ? exp(e) : 0.
//   - Row sums deferred: WMMA accumulates unnormalized, epilogue scales by 1/(s+eps).
//   - WMMA: [16 x 144] (LDS) x [144 x 256] (x from global, L2-resident),
//     each wave owns 4 column tiles; A fragment reused across the 4 tiles.
// -----------------------------------------------------------------------------
__global__ void __launch_bounds__(128)
banded_attn_kernel(const float* __restrict__ x,
                   const float* __restrict__ qbuf,
                   const float* __restrict__ kbuf,
                   const float* __restrict__ bh,
                   const float* __restrict__ Wa,
                   const float* __restrict__ ba,
                   float* __restrict__ out) {
    __shared__ float q_sh[16][33];
    __shared__ float k_sh[JT][33];
    __shared__ float a_sh[16][148];
    __shared__ float bh_sh[UNITS];
    __shared__ float wa_sh[UNITS];
    __shared__ float inv_s[16];

    const int tid = threadIdx.x;
    const int b   = blockIdx.x / (LSEQ / 16);
    const int i0  = (blockIdx.x % (LSEQ / 16)) * 16;
    const int jlo = i0 - 64;

    const float* xb = x + (size_t)b * LSEQ * DDIM;

    if (tid < UNITS) {
        bh_sh[tid] = bh[tid];
        wa_sh[tid] = Wa[tid];
    }
    const float ba0 = ba[0];

    // Stage q rows (16 x 32)
    for (int p = tid; p < 16 * UNITS; p += 128) {
        const int r = p >> 5, u = p & 31;
        q_sh[r][u] = qbuf[((size_t)b * LSEQ + i0 + r) * UNITS + u];
    }
    // Stage k window (144 x 32), clamped at sequence edges (masked out later)
    for (int p = tid; p < JT * UNITS; p += 128) {
        const int jj = p >> 5, u = p & 31;
        const int jc = clampi(jlo + jj, 0, LSEQ - 1);
        k_sh[jj][u] = kbuf[((size_t)b * LSEQ + jc) * UNITS + u];
    }
    // Warm L2->WGP path for the x window used by the WMMA phase
    for (int jj = tid; jj < JT; jj += 128) {
        const int jc = clampi(jlo + jj, 0, LSEQ - 1);
        __builtin_prefetch(&xb[(size_t)jc * DDIM], 0, 3);
    }
    __syncthreads();

    // Emission over the 16 x 144 shifted band tile.
    // Band for row r: jj in [r, r+127]; global clip j in [0, L).
    for (int p = tid; p < 16 * JT; p += 128) {
        const int r  = p / JT;
        const int jj = p - r * JT;
        const int j  = jlo + jj;
        const bool valid = (j >= 0) && (j < LSEQ) && (jj >= r) && (jj <= r + 127);
        float e = ba0;
#pragma unroll 8
        for (int u = 0; u < UNITS; ++u) {
            e += wa_sh[u] * tanhf(q_sh[r][u] + k_sh[jj][u] + bh_sh[u]);
        }
        a_sh[r][jj] = valid ? __expf(e) : 0.0f;
    }
    __syncthreads();

    // Row sums -> deferred normalization factor
    if (tid < 16) {
        float s = 0.0f;
        for (int jj = 0; jj < JT; ++jj) s += a_sh[tid][jj];
        inv_s[tid] = 1.0f / (s + EPSV);
    }
    __syncthreads();

    // WMMA phase: v(16 x 256) = a(16 x 144) @ x_window(144 x 256)
    const int wave = tid >> 5;
    const int lane = tid & 31;
    const int half = lane >> 4;
    const int lr   = lane & 15;

    v8f acc0 = {}, acc1 = {}, acc2 = {}, acc3 = {};
    const int n0 = wave * 64;  // this wave covers columns [n0, n0+64)

    for (int kk = 0; kk < JT; kk += 4) {
        // A fragment from LDS: lanes 0-15 -> K kk+0/1, lanes 16-31 -> kk+2/3
        v2f a;
        a.x = a_sh[lr][kk + half * 2];
        a.y = a_sh[lr][kk + half * 2 + 1];

        const int j0  = jlo + kk + half * 2;
        const int jc0 = clampi(j0, 0, LSEQ - 1);
        const int jc1 = clampi(j0 + 1, 0, LSEQ - 1);
        const size_t r0 = (size_t)jc0 * DDIM;
        const size_t r1 = (size_t)jc1 * DDIM;

        v2f bfrag;
        bfrag.x = xb[r0 + n0 + lr];
        bfrag.y = xb[r1 + n0 + lr];
        acc0 = __builtin_amdgcn_wmma_f32_16x16x4_f32(false, a, false, bfrag, (short)0, acc0, false, false);

        bfrag.x = xb[r0 + n0 + 16 + lr];
        bfrag.y = xb[r1 + n0 + 16 + lr];
        acc1 = __builtin_amdgcn_wmma_f32_16x16x4_f32(false, a, false, bfrag, (short)0, acc1, false, false);

        bfrag.x = xb[r0 + n0 + 32 + lr];
        bfrag.y = xb[r1 + n0 + 32 + lr];
        acc2 = __builtin_amdgcn_wmma_f32_16x16x4_f32(false, a, false, bfrag, (short)0, acc2, false, false);

        bfrag.x = xb[r0 + n0 + 48 + lr];
        bfrag.y = xb[r1 + n0 + 48 + lr];
        acc3 = __builtin_amdgcn_wmma_f32_16x16x4_f32(false, a, false, bfrag, (short)0, acc3, false, false);
    }

    // Epilogue: scale rows by 1/(sum+eps) and store
    for (int v = 0; v < 8; ++v) {
        const int row = v + half * 8;
        const float sc = inv_s[row];
        const size_t base = ((size_t)b * LSEQ + i0 + row) * DDIM + n0;
        out[base + lr]      = acc0[v] * sc;
        out[base + 16 + lr] = acc1[v] * sc;
        out[base + 32 + lr] = acc2[v] * sc;
        out[base + 48 + lr] = acc3[v] * sc;
    }
}

extern "C" void kernel_launch(void* const* d_in, const int* in_sizes, int n_in,
                              void* d_out, int out_size, void* d_ws, size_t ws_size,
                              hipStream_t stream) {
    const float* x  = (const float*)d_in[0];
    const float* Wt = (const float*)d_in[1];
    const float* Wx = (const float*)d_in[2];
    const float* bh = (const float*)d_in[3];
    const float* Wa = (const float*)d_in[4];
    const float* ba = (const float*)d_in[5];
    float* out = (float*)d_out;

    // Workspace: q and k projections, B*L*UNITS floats each (512 KB total)
    float* qbuf = (float*)d_ws;
    float* kbuf = qbuf + (size_t)BATCH * LSEQ * UNITS;

    const int nblk = BATCH * (LSEQ / 16);   // 128 blocks

    qk_proj_kernel<<<nblk, 32, 0, stream>>>(x, Wt, Wx, qbuf, kbuf);
    banded_attn_kernel<<<nblk, 128, 0, stream>>>(x, qbuf, kbuf, bh, Wa, ba, out);
}